// dwloss_56642028700424
// MI455X (gfx1250) — compile-verified
//
#include <hip/hip_runtime.h>
#include <hip/hip_bf16.h>
#include <math.h>

// ---------------------------------------------------------------------------
// Problem constants (from reference): B=64, C=3, H=W=256
// ---------------------------------------------------------------------------
#define BATCH     64
#define SPATIAL   196608          // 3*256*256 (flattened feature dim)
#define GW_EPS    0.0005f
#define OUTER_IT  30
#define INNER_IT  200
#define INV_TEMP  10.0f           // 1 / SOFTMAX_TEMP
#define LOGP      (-4.1588830833596715f)   // log(1/64)

// workspace layout (float offsets)
#define WS_G1    0                // 64x64 gram(lr)
#define WS_G2    4096             // 64x64 gram(sr)
#define WS_C1    8192             // 64x64 normalized cdist(lr)
#define WS_C2    12288            // 64x64 normalized cdist(sr)
#define WS_CC1   16384            // 64   (c1^2)@p
#define WS_CC2   16448            // 64   (c2^2)@q
#define WS_GW    16512
#define WS_KL1   16513
#define WS_KL2   16514
#define WS_ADV   16515
#define WS_FLOATS 16516

typedef __attribute__((ext_vector_type(2))) float v2f;
typedef __attribute__((ext_vector_type(8))) float v8f;

// ---------------------------------------------------------------------------
// 0) zero the accumulator region of the workspace
// ---------------------------------------------------------------------------
__global__ void gwk_init(float* __restrict__ ws) {
    int i = blockIdx.x * blockDim.x + threadIdx.x;
    if (i < WS_FLOATS) ws[i] = 0.0f;
}

// ---------------------------------------------------------------------------
// 1) Gram matrices G = X @ X^T  (64 x 196608 rows) via V_WMMA_F32_16X16X4_F32
//    grid = (GRAM_BLOCKS, 2): y selects lr->G1 / sr->G2; x is a split-K chunk.
//    Tile staging uses float4 (global_load_b128 -> ds_store_b128); LDS row
//    stride 68 floats = 272 B keeps every row 16-B aligned and leaves the
//    WMMA fragment reads bank-conflict-free (lanes 0-15: banks {4L+ka},
//    lanes 16-31: {4L+ka+2}, disjoint).
// ---------------------------------------------------------------------------
#define GRAM_BLOCKS 512
#define BK 64
#define TSA 68                     // gram LDS stride (16B-aligned rows)
#define TSI 65                     // sinkhorn LDS stride (odd: conflict-free sweeps)

__global__ void gwk_gram(const float* __restrict__ lr,
                         const float* __restrict__ sr,
                         float* __restrict__ ws) {
    __shared__ float tile[64 * TSA];

    const float* src = (blockIdx.y == 0) ? lr : sr;
    float* G = ws + (blockIdx.y == 0 ? WS_G1 : WS_G2);

    const int kchunk = SPATIAL / GRAM_BLOCKS;          // 384
    const int kbase0 = blockIdx.x * kchunk;

    const int tid   = threadIdx.x;
    const int wave  = tid >> 5;                        // 0..7
    const int lane  = tid & 31;
    const int lhalf = lane >> 4;                       // 0|1
    const int lmod  = lane & 15;

    // two 16x16 tiles per wave out of the 4x4 tile grid
    const int t0 = wave, t1 = wave + 8;
    const int tm0 = t0 >> 2, tn0 = t0 & 3;
    const int tm1 = t1 >> 2, tn1 = t1 & 3;

    v8f acc0 = {};
    v8f acc1 = {};

    for (int kb = 0; kb < kchunk; kb += BK) {
        const int kbase = kbase0 + kb;
        // cooperative coalesced load: 64 rows x 64 K as float4 (b128)
        #pragma unroll
        for (int i = 0; i < (64 * BK / 4) / 256; ++i) {   // 4 float4 per thread
            int idx4 = tid + i * 256;
            int row  = idx4 >> 4;                         // 16 float4 per row
            int k4   = idx4 & 15;
            const float4 v = *(const float4*)&src[(size_t)row * SPATIAL + kbase + k4 * 4];
            *(float4*)&tile[row * TSA + k4 * 4] = v;
            if (kb + BK < kchunk)   // prefetch next chunk (global_prefetch_b8)
                __builtin_prefetch(&src[(size_t)row * SPATIAL + kbase + BK + k4 * 4], 0, 3);
        }
        __syncthreads();

        #pragma unroll
        for (int k0 = 0; k0 < BK; k0 += 4) {
            const int ka = k0 + 2 * lhalf;             // ISA A-frag: K = k0 + 2*half + vgpr
            v2f a0, b0, a1, b1;
            const int ar0 = (tm0 * 16 + lmod) * TSA;
            const int br0 = (tn0 * 16 + lmod) * TSA;
            a0.x = tile[ar0 + ka]; a0.y = tile[ar0 + ka + 1];
            b0.x = tile[br0 + ka]; b0.y = tile[br0 + ka + 1];
            acc0 = __builtin_amdgcn_wmma_f32_16x16x4_f32(
                false, a0, false, b0, (short)0, acc0, false, false);

            const int ar1 = (tm1 * 16 + lmod) * TSA;
            const int br1 = (tn1 * 16 + lmod) * TSA;
            a1.x = tile[ar1 + ka]; a1.y = tile[ar1 + ka + 1];
            b1.x = tile[br1 + ka]; b1.y = tile[br1 + ka + 1];
            acc1 = __builtin_amdgcn_wmma_f32_16x16x4_f32(
                false, a1, false, b1, (short)0, acc1, false, false);
        }
        __syncthreads();
    }

    // C/D layout: VGPR r -> M = r + 8*(lane>=16), N = lane%16
    #pragma unroll
    for (int r = 0; r < 8; ++r) {
        int row0 = tm0 * 16 + r + 8 * lhalf;
        int col0 = tn0 * 16 + lmod;
        atomicAdd(&G[row0 * 64 + col0], acc0[r]);
        int row1 = tm1 * 16 + r + 8 * lhalf;
        int col1 = tn1 * 16 + lmod;
        atomicAdd(&G[row1 * 64 + col1], acc1[r]);
    }
}

// ---------------------------------------------------------------------------
// 2) From Gram -> normalized distance matrices c1,c2 and constC row/col vecs.
// ---------------------------------------------------------------------------
__global__ void gwk_prep(float* __restrict__ ws) {
    __shared__ float sq[64];
    __shared__ float red[256];
    const int tid = threadIdx.x;

    for (int mat = 0; mat < 2; ++mat) {
        const float* G = ws + (mat == 0 ? WS_G1 : WS_G2);
        float* C  = ws + (mat == 0 ? WS_C1 : WS_C2);
        float* CC = ws + (mat == 0 ? WS_CC1 : WS_CC2);

        if (tid < 64) sq[tid] = G[tid * 64 + tid];
        __syncthreads();

        float cv[16];
        float mloc = 0.0f;
        const int base = tid * 16;                 // row = tid/4, cols contiguous
        const int row  = base >> 6;
        #pragma unroll
        for (int i = 0; i < 16; ++i) {
            int col = (base + i) & 63;
            float d2 = sq[row] + sq[col] - 2.0f * G[base + i];
            float d  = sqrtf(fmaxf(d2, 0.0f));
            cv[i] = d;
            mloc = fmaxf(mloc, d);
        }
        red[tid] = mloc;
        __syncthreads();
        for (int s = 128; s > 0; s >>= 1) {
            if (tid < s) red[tid] = fmaxf(red[tid], red[tid + s]);
            __syncthreads();
        }
        float inv = (red[0] > 0.0f) ? (1.0f / red[0]) : 1.0f;
        __syncthreads();

        float ssq = 0.0f;
        #pragma unroll
        for (int i = 0; i < 16; ++i) {
            float c = cv[i] * inv;
            C[base + i] = c;
            ssq += c * c;
        }
        red[tid] = ssq;
        __syncthreads();
        if (tid < 64) {
            float s = red[4 * tid] + red[4 * tid + 1] + red[4 * tid + 2] + red[4 * tid + 3];
            CC[tid] = s * (1.0f / 64.0f);          // (c^2) @ p, p = 1/64
        }
        __syncthreads();
    }
}

// ---------------------------------------------------------------------------
// 3) Entropic GW: 30 x (tens(T) -> 200 log-Sinkhorn sweeps) on 64x64 in LDS.
//    Odd stride 65 => conflict-free row AND column sweeps on 64 banks.
// ---------------------------------------------------------------------------
__global__ void gwk_iter(float* __restrict__ ws) {
    __shared__ float T  [64 * TSI];
    __shared__ float Mk [64 * TSI];
    __shared__ float tmp[64 * TSI];
    __shared__ float av[64], bv[64], cc1s[64], cc2s[64];
    __shared__ float red[256];

    const float* C1 = ws + WS_C1;
    const float* C2 = ws + WS_C2;
    const int tid = threadIdx.x;

    if (tid < 64) { cc1s[tid] = ws[WS_CC1 + tid]; cc2s[tid] = ws[WS_CC2 + tid]; }
    for (int idx = tid; idx < 4096; idx += 256)
        T[(idx >> 6) * TSI + (idx & 63)] = 1.0f / 4096.0f;   // p q^T
    __syncthreads();

    for (int outer = 0; outer < OUTER_IT; ++outer) {
        // tmp = T @ (2*C2)^T : tmp[i][j] = 2 * sum_l T[i][l] * C2[j][l]
        for (int idx = tid; idx < 4096; idx += 256) {
            int i = idx >> 6, j = idx & 63;
            float s = 0.0f;
            for (int l = 0; l < 64; ++l) s += T[i * TSI + l] * C2[j * 64 + l];
            tmp[i * TSI + j] = 2.0f * s;
        }
        __syncthreads();
        // Mk = -(constC - C1@tmp)/eps
        for (int idx = tid; idx < 4096; idx += 256) {
            int i = idx >> 6, j = idx & 63;
            float s = 0.0f;
            for (int k = 0; k < 64; ++k) s += C1[i * 64 + k] * tmp[k * TSI + j];
            Mk[i * TSI + j] = (s - cc1s[i] - cc2s[j]) * (1.0f / GW_EPS);
        }
        if (tid < 64) { av[tid] = 0.0f; bv[tid] = 0.0f; }
        __syncthreads();

        for (int it = 0; it < INNER_IT; ++it) {
            if (tid < 64) {                           // a = logp - lse_j(Mk + b)
                float m = -INFINITY;
                for (int j = 0; j < 64; ++j) m = fmaxf(m, Mk[tid * TSI + j] + bv[j]);
                float s = 0.0f;
                for (int j = 0; j < 64; ++j) s += __expf(Mk[tid * TSI + j] + bv[j] - m);
                av[tid] = LOGP - (m + __logf(s));
            }
            __syncthreads();
            if (tid < 64) {                           // b = logq - lse_i(Mk + a)
                float m = -INFINITY;
                for (int i = 0; i < 64; ++i) m = fmaxf(m, Mk[i * TSI + tid] + av[i]);
                float s = 0.0f;
                for (int i = 0; i < 64; ++i) s += __expf(Mk[i * TSI + tid] + av[i] - m);
                bv[tid] = LOGP - (m + __logf(s));
            }
            __syncthreads();
        }
        for (int idx = tid; idx < 4096; idx += 256) {
            int i = idx >> 6, j = idx & 63;
            T[i * TSI + j] = __expf(av[i] + Mk[i * TSI + j] + bv[j]);
        }
        __syncthreads();
    }

    // loss = sum(tens(T) * T) with the final T
    for (int idx = tid; idx < 4096; idx += 256) {
        int i = idx >> 6, j = idx & 63;
        float s = 0.0f;
        for (int l = 0; l < 64; ++l) s += T[i * TSI + l] * C2[j * 64 + l];
        tmp[i * TSI + j] = 2.0f * s;
    }
    __syncthreads();
    float acc = 0.0f;
    for (int idx = tid; idx < 4096; idx += 256) {
        int i = idx >> 6, j = idx & 63;
        float s = 0.0f;
        for (int k = 0; k < 64; ++k) s += C1[i * 64 + k] * tmp[k * TSI + j];
        acc += (cc1s[i] + cc2s[j] - s) * T[i * TSI + j];
    }
    red[tid] = acc;
    __syncthreads();
    for (int s = 128; s > 0; s >>= 1) {
        if (tid < s) red[tid] += red[tid + s];
        __syncthreads();
    }
    if (tid == 0) ws[WS_GW] = red[0];
}

// ---------------------------------------------------------------------------
// 4) JS divergence: batch-dim softmax over 64 per column; 196608 columns.
//    4 columns per thread via float4 (global_load_b128), coalesced.
// ---------------------------------------------------------------------------
#define SP4 (SPATIAL / 4)          // 49152 float4 columns

__global__ void gwk_js(const float* __restrict__ sr,
                       const float* __restrict__ hr,
                       float* __restrict__ ws) {
    __shared__ float r1[256], r2[256];
    const int c4 = blockIdx.x * blockDim.x + threadIdx.x;
    float kl1 = 0.0f, kl2 = 0.0f;
    if (c4 < SP4) {
        const float4* x4 = (const float4*)sr;
        const float4* y4 = (const float4*)hr;
        float4 mx = make_float4(-INFINITY, -INFINITY, -INFINITY, -INFINITY);
        float4 my = mx;
        for (int b = 0; b < BATCH; ++b) {
            float4 xv = x4[(size_t)b * SP4 + c4];
            float4 yv = y4[(size_t)b * SP4 + c4];
            mx.x = fmaxf(mx.x, xv.x); mx.y = fmaxf(mx.y, xv.y);
            mx.z = fmaxf(mx.z, xv.z); mx.w = fmaxf(mx.w, xv.w);
            my.x = fmaxf(my.x, yv.x); my.y = fmaxf(my.y, yv.y);
            my.z = fmaxf(my.z, yv.z); my.w = fmaxf(my.w, yv.w);
        }
        float4 sx = make_float4(0.f, 0.f, 0.f, 0.f);
        float4 sy = sx;
        for (int b = 0; b < BATCH; ++b) {
            float4 xv = x4[(size_t)b * SP4 + c4];
            float4 yv = y4[(size_t)b * SP4 + c4];
            sx.x += __expf(xv.x - mx.x); sx.y += __expf(xv.y - mx.y);
            sx.z += __expf(xv.z - mx.z); sx.w += __expf(xv.w - mx.w);
            sy.x += __expf(yv.x - my.x); sy.y += __expf(yv.y - my.y);
            sy.z += __expf(yv.z - my.z); sy.w += __expf(yv.w - my.w);
        }
        float4 lsx = make_float4(__logf(sx.x), __logf(sx.y), __logf(sx.z), __logf(sx.w));
        float4 lsy = make_float4(__logf(sy.x), __logf(sy.y), __logf(sy.z), __logf(sy.w));
        for (int b = 0; b < BATCH; ++b) {
            float4 xv = x4[(size_t)b * SP4 + c4];
            float4 yv = y4[(size_t)b * SP4 + c4];
            #pragma unroll
            for (int q = 0; q < 4; ++q) {
                float xq = (&xv.x)[q], yq = (&yv.x)[q];
                float lx = xq - (&mx.x)[q] - (&lsx.x)[q];     // log_softmax
                float ly = yq - (&my.x)[q] - (&lsy.x)[q];
                float px = __expf(lx), py = __expf(ly);
                float m  = 0.5f * (px + py);
                float lm = __logf(m);
                kl1 += m * (lm - lx);
                kl2 += m * (lm - ly);
            }
        }
    }
    r1[threadIdx.x] = kl1; r2[threadIdx.x] = kl2;
    __syncthreads();
    for (int s = 128; s > 0; s >>= 1) {
        if (threadIdx.x < s) {
            r1[threadIdx.x] += r1[threadIdx.x + s];
            r2[threadIdx.x] += r2[threadIdx.x + s];
        }
        __syncthreads();
    }
    if (threadIdx.x == 0) {
        atomicAdd(&ws[WS_KL1], r1[0]);
        atomicAdd(&ws[WS_KL2], r2[0]);
    }
}

// ---------------------------------------------------------------------------
// 5) adversarial term: sum softplus(-x) over 64*1*16*16 = 16384 elements
// ---------------------------------------------------------------------------
__global__ void gwk_adv(const float* __restrict__ disc, float* __restrict__ ws) {
    __shared__ float red[256];
    const int i = blockIdx.x * blockDim.x + threadIdx.x;
    float v = 0.0f;
    if (i < 16384) {
        float z = -disc[i];
        v = fmaxf(z, 0.0f) + log1pf(__expf(-fabsf(z)));   // stable softplus
    }
    red[threadIdx.x] = v;
    __syncthreads();
    for (int s = 128; s > 0; s >>= 1) {
        if (threadIdx.x < s) red[threadIdx.x] += red[threadIdx.x + s];
        __syncthreads();
    }
    if (threadIdx.x == 0) atomicAdd(&ws[WS_ADV], red[0]);
}

// ---------------------------------------------------------------------------
// 6) softmax-weighted combine -> scalar output
// ---------------------------------------------------------------------------
__global__ void gwk_final(const float* __restrict__ ws, float* __restrict__ out) {
    float gw  = ws[WS_GW];
    float js  = 0.5f * (ws[WS_KL1] + ws[WS_KL2]) * (1.0f / (float)BATCH);
    float adv = ws[WS_ADV] * (1.0f / 16384.0f);
    float l0 = gw * INV_TEMP, l1 = js * INV_TEMP, l2 = adv * INV_TEMP;
    float mm = fmaxf(l0, fmaxf(l1, l2));
    float e0 = __expf(l0 - mm), e1 = __expf(l1 - mm), e2 = __expf(l2 - mm);
    float inv = 1.0f / (e0 + e1 + e2);
    out[0] = (gw * e0 + js * e1 + adv * e2) * inv;
}

// ---------------------------------------------------------------------------
extern "C" void kernel_launch(void* const* d_in, const int* in_sizes, int n_in,
                              void* d_out, int out_size, void* d_ws, size_t ws_size,
                              hipStream_t stream) {
    const float* lr   = (const float*)d_in[0];
    const float* sr   = (const float*)d_in[1];
    const float* hr   = (const float*)d_in[2];
    const float* disc = (const float*)d_in[3];
    float* out = (float*)d_out;
    float* ws  = (float*)d_ws;

    gwk_init<<<(WS_FLOATS + 255) / 256, 256, 0, stream>>>(ws);
    gwk_gram<<<dim3(GRAM_BLOCKS, 2), 256, 0, stream>>>(lr, sr, ws);
    gwk_prep<<<1, 256, 0, stream>>>(ws);
    gwk_iter<<<1, 256, 0, stream>>>(ws);
    gwk_js<<<(SP4 + 255) / 256, 256, 0, stream>>>(sr, hr, ws);
    gwk_adv<<<64, 256, 0, stream>>>(disc, ws);
    gwk_final<<<1, 1, 0, stream>>>(ws, out);
}